// GCU_1305670058421
// MI455X (gfx1250) — compile-verified
//
#include <hip/hip_runtime.h>
#include <hip/hip_bf16.h>

// ---------------------------------------------------------------------------
// Problem constants (from reference): B=16, HT=WD=64 -> HW=4096, D=2048,
// V=32 clusters, F=128 out features.
// ---------------------------------------------------------------------------
#define BB  16
#define HW  4096
#define DD  2048
#define VV  32
#define FF  128

typedef float v2f __attribute__((ext_vector_type(2)));
typedef float v8f __attribute__((ext_vector_type(8)));

// f32 WMMA 16x16x4: D(16x16 f32) = A(16x4 f32) * B(4x16 f32) + C
// 8-arg form: (neg_a, A, neg_b, B, c_mod, C, reuse_a, reuse_b)
#define WMMA4(a, b, c) \
  __builtin_amdgcn_wmma_f32_16x16x4_f32(false, (a), false, (b), (short)0, (c), false, false)

// Workspace layout (float offsets)
#define OFF_Q     ((size_t)0)            // B*HW*V   = 2097152
#define OFF_ZPRE  ((size_t)2097152)      // B*D*V    = 1048576
#define OFF_M1    ((size_t)3145728)      // packed inv^2       = 65536
#define OFF_M2    ((size_t)3211264)      // packed -2*W*inv^2  = 65536
#define OFF_T3    ((size_t)3276800)      // 32
#define OFF_N     ((size_t)3276832)      // B*V = 512  (zeroed)
#define OFF_S     ((size_t)3277344)      // B*V = 512  (zeroed)
#define OFF_P     ((size_t)3277856)      // B*V*F = 65536
#define OFF_G     ((size_t)3343392)      // B*V*V = 16384
#define OFF_ZO    ((size_t)3359776)      // B*V*F = 65536

// ---------------------------------------------------------------------------
// Prep 1: pack B-operands for the Q GEMM into per-lane WMMA order.
// Packed index = ((tile*512 + kstep)*32 + lane)*2 + j, where lane<16 holds
// K rows {4k, 4k+1} and lane>=16 holds {4k+2, 4k+3}, N = 16*tile + (lane&15).
// ---------------------------------------------------------------------------
__global__ __launch_bounds__(256) void k_pack_b(const float* __restrict__ W,
                                                const float* __restrict__ Var,
                                                float* __restrict__ M1sw,
                                                float* __restrict__ M2sw) {
  int gid = blockIdx.x * 256 + threadIdx.x;   // 65536 total
  int j = gid & 1;
  int l = (gid >> 1) & 31;
  int s = (gid >> 6) & 511;
  int t = (gid >> 15) & 1;
  int k = 4 * s + ((l < 16) ? j : (2 + j));
  int v = 16 * t + (l & 15);
  float w  = W[(size_t)k * VV + v];
  float va = Var[(size_t)k * VV + v];
  float inv2 = 1.0f / (va * va);
  M1sw[gid] = inv2;
  M2sw[gid] = -2.0f * w * inv2;
}

// Prep 2: T3[v] = sum_c W^2 * inv^2
__global__ __launch_bounds__(256) void k_t3(const float* __restrict__ W,
                                            const float* __restrict__ Var,
                                            float* __restrict__ T3) {
  __shared__ float red[256];
  int v = blockIdx.x;
  float acc = 0.0f;
  for (int c = threadIdx.x; c < DD; c += 256) {
    float w  = W[(size_t)c * VV + v];
    float va = Var[(size_t)c * VV + v];
    acc += (w * w) / (va * va);
  }
  red[threadIdx.x] = acc;
  __syncthreads();
  for (int s = 128; s > 0; s >>= 1) {
    if (threadIdx.x < s) red[threadIdx.x] += red[threadIdx.x + s];
    __syncthreads();
  }
  if (threadIdx.x == 0) T3[v] = red[0];
}

// ---------------------------------------------------------------------------
// Stage 1: soft-assignment Q (B,HW,V).  One wave = 16 rows x 32 clusters.
// Q_pre = X^2 @ inv2 + X @ (-2 W inv2) + T3  -> rowmin/exp/normalize fused.
// Also accumulates n[b,v] = sum_p Q via global f32 atomics.
// ---------------------------------------------------------------------------
__global__ __launch_bounds__(256) void k_qassign(const float* __restrict__ X,
                                                 const float* __restrict__ M1sw,
                                                 const float* __restrict__ M2sw,
                                                 const float* __restrict__ T3,
                                                 float* __restrict__ Q,
                                                 float* __restrict__ nBV) {
  const int lane = threadIdx.x & 31;
  const int wave = threadIdx.x >> 5;
  const int wg   = blockIdx.x * 8 + wave;     // 4096 waves
  const int b    = wg >> 8;
  const int row0 = (wg & 255) << 4;
  const int half = lane >> 4;
  const int lm   = lane & 15;

  const float* pA  = X + (size_t)(b * HW + row0 + lm) * DD + 2 * half;
  const float* pB1 = M1sw + lane * 2;
  const float* pB2 = M2sw + lane * 2;

  v8f c0 = {0, 0, 0, 0, 0, 0, 0, 0};
  v8f c1 = {0, 0, 0, 0, 0, 0, 0, 0};

#pragma unroll 2
  for (int s = 0; s < 512; ++s) {
    v2f a   = *(const v2f*)(pA + 4 * s);
    v2f a2  = a * a;
    v2f b10 = *(const v2f*)(pB1 + s * 64);
    v2f b11 = *(const v2f*)(pB1 + 32768 + s * 64);
    v2f b20 = *(const v2f*)(pB2 + s * 64);
    v2f b21 = *(const v2f*)(pB2 + 32768 + s * 64);
    c0 = WMMA4(a2, b10, c0);
    c0 = WMMA4(a,  b20, c0);
    c1 = WMMA4(a2, b11, c1);
    c1 = WMMA4(a,  b21, c1);
  }

  const float t30 = T3[lm];
  const float t31 = T3[16 + lm];

  float na0 = 0.0f, na1 = 0.0f;
#pragma unroll
  for (int r = 0; r < 8; ++r) {
    float q0 = c0[r] + t30;
    float q1 = c1[r] + t31;
    // row-min over 32 clusters: pair-min then 16-lane xor reduction
    float mn = fminf(q0, q1);
    mn = fminf(mn, __shfl_xor(mn, 1, 32));
    mn = fminf(mn, __shfl_xor(mn, 2, 32));
    mn = fminf(mn, __shfl_xor(mn, 4, 32));
    mn = fminf(mn, __shfl_xor(mn, 8, 32));
    q0 = __expf(-0.5f * (q0 - mn));
    q1 = __expf(-0.5f * (q1 - mn));
    float ss = q0 + q1;
    ss += __shfl_xor(ss, 1, 32);
    ss += __shfl_xor(ss, 2, 32);
    ss += __shfl_xor(ss, 4, 32);
    ss += __shfl_xor(ss, 8, 32);
    float inv = 1.0f / ss;
    q0 *= inv;
    q1 *= inv;
    size_t o = (size_t)(b * HW + row0 + r + 8 * half) * VV + lm;
    Q[o]      = q0;
    Q[o + 16] = q1;
    na0 += q0;
    na1 += q1;
  }
  na0 += __shfl_xor(na0, 16, 32);
  na1 += __shfl_xor(na1, 16, 32);
  if (lane < 16) {
    atomicAdd(&nBV[b * VV + lm], na0);
    atomicAdd(&nBV[b * VV + 16 + lm], na1);
  }
}

// ---------------------------------------------------------------------------
// Stage 2: XtQ[b,c,v] = sum_p X[b,p,c] Q[b,p,v] (second X pass), then
// Zpre = (XtQ - W*n) * inv / n, plus atomic sum of Zpre^2 per (b,v).
// ---------------------------------------------------------------------------
__global__ __launch_bounds__(256) void k_xtq(const float* __restrict__ X,
                                             const float* __restrict__ Qm,
                                             const float* __restrict__ W,
                                             const float* __restrict__ Var,
                                             const float* __restrict__ nBV,
                                             float* __restrict__ Zpre,
                                             float* __restrict__ Ssq) {
  const int lane = threadIdx.x & 31;
  const int wave = threadIdx.x >> 5;
  const int wg   = blockIdx.x * 8 + wave;     // 2048 waves
  const int b    = wg >> 7;
  const int c0   = (wg & 127) << 4;
  const int half = lane >> 4;
  const int lm   = lane & 15;

  const float*  pX = X  + (size_t)b * HW * DD + (c0 + lm);
  const float*  pQ = Qm + (size_t)b * HW * VV + lm;
  const size_t rA0 = (size_t)(2 * half) * DD;
  const size_t rA1 = rA0 + DD;
  const int    rB0 = (2 * half) * VV;
  const int    rB1 = rB0 + VV;

  v8f c0acc = {0, 0, 0, 0, 0, 0, 0, 0};
  v8f c1acc = {0, 0, 0, 0, 0, 0, 0, 0};

#pragma unroll 2
  for (int s = 0; s < 1024; ++s) {
    const float* xk = pX + (size_t)(4 * s) * DD;
    const float* qk = pQ + 4 * s * VV;
    v2f a;  a.x  = xk[rA0];       a.y  = xk[rA1];
    v2f b0; b0.x = qk[rB0];       b0.y = qk[rB1];
    v2f b1; b1.x = qk[rB0 + 16];  b1.y = qk[rB1 + 16];
    c0acc = WMMA4(a, b0, c0acc);
    c1acc = WMMA4(a, b1, c1acc);
  }

  const float nv0 = nBV[b * VV + lm];
  const float nv1 = nBV[b * VV + 16 + lm];

  float s0 = 0.0f, s1 = 0.0f;
#pragma unroll
  for (int r = 0; r < 8; ++r) {
    int cidx = c0 + r + 8 * half;
    size_t pb = (size_t)cidx * VV;
    float w0 = W[pb + lm],      w1 = W[pb + 16 + lm];
    float v0 = Var[pb + lm],    v1 = Var[pb + 16 + lm];
    float z0 = (c0acc[r] - w0 * nv0) / (v0 * nv0);
    float z1 = (c1acc[r] - w1 * nv1) / (v1 * nv1);
    size_t o = (size_t)(b * DD + cidx) * VV;
    Zpre[o + lm]      = z0;
    Zpre[o + 16 + lm] = z1;
    s0 += z0 * z0;
    s1 += z1 * z1;
  }
  s0 += __shfl_xor(s0, 16, 32);
  s1 += __shfl_xor(s1, 16, 32);
  if (lane < 16) {
    atomicAdd(&Ssq[b * VV + lm], s0);
    atomicAdd(&Ssq[b * VV + 16 + lm], s1);
  }
}

// ---------------------------------------------------------------------------
// Stage 3: P = Zpre^T @ weight (32x128) and G = Zpre^T @ Zpre (32x32) per b.
// 10 N-tiles per (b, mtile): tiles 0..7 -> weight, 8..9 -> Zpre (Gram).
// ---------------------------------------------------------------------------
__global__ __launch_bounds__(256) void k_pg(const float* __restrict__ Zpre,
                                            const float* __restrict__ weight,
                                            float* __restrict__ P,
                                            float* __restrict__ G) {
  const int lane = threadIdx.x & 31;
  const int wave = threadIdx.x >> 5;
  const int wg   = blockIdx.x * 8 + wave;     // 320 waves
  const int b    = wg / 20;
  const int rem  = wg % 20;
  const int mt   = rem / 10;
  const int nt   = rem % 10;
  const int half = lane >> 4;
  const int lm   = lane & 15;

  const float* pA = Zpre + (size_t)b * DD * VV + 16 * mt + lm;
  const float* pB;
  int strideB;
  if (nt < 8) { pB = weight + 16 * nt + lm;                          strideB = FF; }
  else        { pB = Zpre + (size_t)b * DD * VV + 16 * (nt - 8) + lm; strideB = VV; }

  v8f acc = {0, 0, 0, 0, 0, 0, 0, 0};
#pragma unroll 2
  for (int s = 0; s < 512; ++s) {
    int k = 4 * s + 2 * half;
    v2f a;  a.x = pA[(size_t)k * VV];       a.y = pA[(size_t)(k + 1) * VV];
    v2f bb; bb.x = pB[(size_t)k * strideB]; bb.y = pB[(size_t)(k + 1) * strideB];
    acc = WMMA4(a, bb, acc);
  }

#pragma unroll
  for (int r = 0; r < 8; ++r) {
    int m = 16 * mt + r + 8 * half;
    if (nt < 8) P[((size_t)b * VV + m) * FF + 16 * nt + lm] = acc[r];
    else        G[((size_t)b * VV + m) * VV + 16 * (nt - 8) + lm] = acc[r];
  }
}

// ---------------------------------------------------------------------------
// Stage 4: Adj = diag(1/S) G diag(1/S); out1 = diag(1/S) P;
// Zo = relu(Adj @ out1).  Small (32x32x128 per b): LDS + VALU.
// ---------------------------------------------------------------------------
__global__ __launch_bounds__(128) void k_adj(const float* __restrict__ G,
                                             const float* __restrict__ P,
                                             const float* __restrict__ Ssq,
                                             float* __restrict__ Zo) {
  __shared__ float sAdj[VV * VV];
  __shared__ float sS[VV];
  const int b = blockIdx.x;
  const int f = threadIdx.x;

  if (f < VV) sS[f] = Ssq[b * VV + f];
  __syncthreads();
#pragma unroll
  for (int i = 0; i < 8; ++i) {
    int idx = f + i * 128;
    int u = idx >> 5, v = idx & 31;
    sAdj[idx] = G[(size_t)b * VV * VV + idx] / (sS[u] * sS[v]);
  }
  __syncthreads();

  float pv[VV];
#pragma unroll
  for (int u = 0; u < VV; ++u)
    pv[u] = P[((size_t)b * VV + u) * FF + f] / sS[u];

  for (int v = 0; v < VV; ++v) {
    float acc = 0.0f;
#pragma unroll
    for (int u = 0; u < VV; ++u) acc += sAdj[v * VV + u] * pv[u];
    Zo[((size_t)b * VV + v) * FF + f] = fmaxf(acc, 0.0f);
  }
}

// ---------------------------------------------------------------------------
// Stage 5: X_new[b,p,f] = Q[b,p,:] @ Zo[b,:,f].  M=HW, N=128, K=32.
// Output written flat (b,p,f) == reference's raw reshape to (B,F,HT,WD).
// ---------------------------------------------------------------------------
__global__ __launch_bounds__(256) void k_final(const float* __restrict__ Qm,
                                               const float* __restrict__ Zo,
                                               float* __restrict__ out) {
  const int lane = threadIdx.x & 31;
  const int wave = threadIdx.x >> 5;
  const int wg   = blockIdx.x * 8 + wave;     // 16384 waves
  const int b    = wg >> 10;
  const int rem  = wg & 1023;
  const int row0 = (rem >> 2) << 4;
  const int n0   = (rem & 3) << 5;
  const int half = lane >> 4;
  const int lm   = lane & 15;

  const float* pA = Qm + (size_t)(b * HW + row0 + lm) * VV + 2 * half;
  const float* pB = Zo + (size_t)b * VV * FF + n0 + lm;

  v8f c0 = {0, 0, 0, 0, 0, 0, 0, 0};
  v8f c1 = {0, 0, 0, 0, 0, 0, 0, 0};
#pragma unroll
  for (int s = 0; s < 8; ++s) {
    int k = 4 * s + 2 * half;
    v2f a = *(const v2f*)(pA + 4 * s);
    v2f b0; b0.x = pB[k * FF];      b0.y = pB[(k + 1) * FF];
    v2f b1; b1.x = pB[k * FF + 16]; b1.y = pB[(k + 1) * FF + 16];
    c0 = WMMA4(a, b0, c0);
    c1 = WMMA4(a, b1, c1);
  }
#pragma unroll
  for (int r = 0; r < 8; ++r) {
    size_t o = (size_t)(b * HW + row0 + r + 8 * half) * FF + n0 + lm;
    out[o]      = c0[r];
    out[o + 16] = c1[r];
  }
}

// ---------------------------------------------------------------------------
extern "C" void kernel_launch(void* const* d_in, const int* in_sizes, int n_in,
                              void* d_out, int out_size, void* d_ws, size_t ws_size,
                              hipStream_t stream) {
  (void)in_sizes; (void)n_in; (void)out_size; (void)ws_size;
  const float* X   = (const float*)d_in[0];
  const float* W   = (const float*)d_in[1];
  const float* Var = (const float*)d_in[2];
  const float* Wt  = (const float*)d_in[3];
  float* out = (float*)d_out;
  float* ws  = (float*)d_ws;

  float* Qb   = ws + OFF_Q;
  float* Zpre = ws + OFF_ZPRE;
  float* M1sw = ws + OFF_M1;
  float* M2sw = ws + OFF_M2;
  float* T3   = ws + OFF_T3;
  float* nBV  = ws + OFF_N;
  float* Ssq  = ws + OFF_S;
  float* P    = ws + OFF_P;
  float* G    = ws + OFF_G;
  float* Zo   = ws + OFF_ZO;

  // zero n[b,v] and S[b,v] (contiguous 1024 floats)
  hipMemsetAsync(nBV, 0, 1024 * sizeof(float), stream);

  k_pack_b<<<256, 256, 0, stream>>>(W, Var, M1sw, M2sw);
  k_t3<<<32, 256, 0, stream>>>(W, Var, T3);
  k_qassign<<<512, 256, 0, stream>>>(X, M1sw, M2sw, T3, Qb, nBV);
  k_xtq<<<256, 256, 0, stream>>>(X, Qb, W, Var, nBV, Zpre, Ssq);
  k_pg<<<40, 256, 0, stream>>>(Zpre, Wt, P, G);
  k_adj<<<BB, 128, 0, stream>>>(G, P, Ssq, Zo);
  k_final<<<2048, 256, 0, stream>>>(Qb, Zo, out);
}